// CenterCrop_4827543240892
// MI455X (gfx1250) — compile-verified
//
#include <hip/hip_runtime.h>

// CenterCrop + bilinear resize, MI455X (gfx1250, wave32).
// Memory-bound (~0.5 FLOP/byte): direct gather kernel, NT stores to keep
// the input working set resident in the 192MB L2.

#define OUT_H 224
#define OUT_W 224
#define H_MAX 512
#define W_STRIDE 513          // input row stride (last column holds h/w metadata)
#define RESIZE_TO 256.0f

__global__ __launch_bounds__(OUT_W) void center_crop_bilinear_kernel(
    const float* __restrict__ x, float* __restrict__ out)
{
    const int ox = threadIdx.x;          // 0..223, fastest dim -> coalesced
    int lin = blockIdx.x;                // ((b*3 + c)*OUT_H + oy)
    const int oy = lin % OUT_H; lin /= OUT_H;
    const int c  = lin % 3;     lin /= 3;
    const int b  = lin;

    // Per-image metadata: x[b,0,0,-1] = h, x[b,1,0,-1] = w  (uniform per block)
    const size_t plane = (size_t)H_MAX * W_STRIDE;
    const float h = x[(size_t)(b * 3 + 0) * plane + (W_STRIDE - 1)];
    const float w = x[(size_t)(b * 3 + 1) * plane + (W_STRIDE - 1)];

    const float min_dim = fminf(h, w);
    const float scale   = RESIZE_TO / min_dim;
    const float h_res   = rintf(h * scale);                 // round-half-even, matches jnp.round
    const float w_res   = rintf(w * scale);
    const float top     = rintf((h_res - (float)OUT_H) * 0.5f);
    const float left    = rintf((w_res - (float)OUT_W) * 0.5f);

    const int hi = (int)h - 1;
    const int wi = (int)w - 1;

    // Vertical sample position (uniform across the block)
    float src_y = ((float)oy + top + 0.5f) * h / h_res - 0.5f;
    src_y = fminf(fmaxf(src_y, 0.0f), h - 1.0f);
    const float y0f = floorf(src_y);
    const float wy  = src_y - y0f;
    int y0 = (int)y0f;
    y0 = y0 < 0 ? 0 : (y0 > hi ? hi : y0);
    const int y1 = (y0 + 1) < hi ? (y0 + 1) : hi;

    // Horizontal sample position (per lane, monotone in ox -> good L0 locality)
    float src_x = ((float)ox + left + 0.5f) * w / w_res - 0.5f;
    src_x = fminf(fmaxf(src_x, 0.0f), w - 1.0f);
    const float x0f = floorf(src_x);
    const float wx  = src_x - x0f;
    int x0 = (int)x0f;
    x0 = x0 < 0 ? 0 : (x0 > wi ? wi : x0);
    const int x1 = (x0 + 1) < wi ? (x0 + 1) : wi;

    const float* __restrict__ imgp = x + (size_t)(b * 3 + c) * plane;
    const float* __restrict__ row0 = imgp + (size_t)y0 * W_STRIDE;
    const float* __restrict__ row1 = imgp + (size_t)y1 * W_STRIDE;

    const float v00 = row0[x0];
    const float v01 = row0[x1];
    const float v10 = row1[x0];
    const float v11 = row1[x1];

    const float omy = 1.0f - wy;
    const float omx = 1.0f - wx;
    // Exact same 4-term form as the reference for bit-closeness
    const float r = v00 * omy * omx + v01 * omy * wx
                  + v10 * wy  * omx + v11 * wy  * wx;

    const size_t o = (((size_t)(b * 3 + c) * OUT_H) + oy) * OUT_W + ox;
    // Output has zero reuse: non-temporal store (TH=NT) keeps the gathered
    // input working set (~156MB) resident in the 192MB L2.
    __builtin_nontemporal_store(r, &out[o]);
}

extern "C" void kernel_launch(void* const* d_in, const int* in_sizes, int n_in,
                              void* d_out, int out_size, void* d_ws, size_t ws_size,
                              hipStream_t stream) {
    const float* xin = (const float*)d_in[0];
    float* out = (float*)d_out;

    // Derive batch from input size: B * 3 * 512 * 513 elements
    const int B = in_sizes[0] / (3 * H_MAX * W_STRIDE);

    const int nblocks = B * 3 * OUT_H;   // one block per (b, c, oy); 224 lanes = 7 waves
    center_crop_bilinear_kernel<<<nblocks, OUT_W, 0, stream>>>(xin, out);
}